// VectorDistance_36258113913184
// MI455X (gfx1250) — compile-verified
//
#include <hip/hip_runtime.h>
#include <hip/hip_bf16.h>
#include <float.h>

// Problem constants (from reference)
#define L_DIM   173          // number of centroids
#define D_DIM   32           // vector dim
#define NB1     2048         // blocks for streaming reduction
#define T1      256          // threads per block (8 waves, wave32)
#define NSHARD  32           // atomic shards for cross-block combine
// ws layout: float[NSHARD][33]  (33 = 32 column sums + 1 sum-of-squares)

typedef __attribute__((ext_vector_type(2))) float v2f;
typedef __attribute__((ext_vector_type(8))) float v8f;

__global__ void zero_ws_kernel(float* __restrict__ ws) {
    for (int i = threadIdx.x; i < NSHARD * 33; i += blockDim.x) ws[i] = 0.0f;
}

// Kernel 1: stream 128MB of x once; produce sharded partials of
//   colsum[0..31] (sum over batch per dim) and sumsq (Frobenius^2).
__global__ __launch_bounds__(T1) void stat_kernel(const float* __restrict__ x,
                                                  float* __restrict__ ws,
                                                  int n4) {
    __shared__ float s_stat[33];
    const int t = threadIdx.x;
    if (t < 33) s_stat[t] = 0.0f;
    __syncthreads();

    const float4* __restrict__ x4 = (const float4*)x;
    const int gtid   = blockIdx.x * T1 + t;
    const int stride = gridDim.x * T1;          // multiple of 8 -> column base fixed
    // This thread always touches columns base..base+3 (row-major, 32 cols).
    float4 acc = make_float4(0.f, 0.f, 0.f, 0.f);
    float  ssq = 0.0f;

    for (int i = gtid; i < n4; i += stride) {
        if (i + stride < n4) __builtin_prefetch(&x4[i + stride], 0, 0); // global_prefetch_b8
        float4 v = x4[i];                                               // global_load_b128
        acc.x += v.x; acc.y += v.y; acc.z += v.z; acc.w += v.w;
        ssq = __fmaf_rn(v.x, v.x, ssq);
        ssq = __fmaf_rn(v.y, v.y, ssq);
        ssq = __fmaf_rn(v.z, v.z, ssq);
        ssq = __fmaf_rn(v.w, v.w, ssq);
    }

    // Wave32 combine: lanes {l, l+8, l+16, l+24} hold the same column set.
    acc.x += __shfl_xor(acc.x, 8, 32);  acc.y += __shfl_xor(acc.y, 8, 32);
    acc.z += __shfl_xor(acc.z, 8, 32);  acc.w += __shfl_xor(acc.w, 8, 32);
    acc.x += __shfl_xor(acc.x, 16, 32); acc.y += __shfl_xor(acc.y, 16, 32);
    acc.z += __shfl_xor(acc.z, 16, 32); acc.w += __shfl_xor(acc.w, 16, 32);
    // Full-wave reduce for sum of squares.
    ssq += __shfl_xor(ssq, 1, 32);
    ssq += __shfl_xor(ssq, 2, 32);
    ssq += __shfl_xor(ssq, 4, 32);
    ssq += __shfl_xor(ssq, 8, 32);
    ssq += __shfl_xor(ssq, 16, 32);

    const int lane = t & 31;
    if (lane < 8) {
        const int base = lane * 4;              // == (gtid % 8) * 4
        atomicAdd(&s_stat[base + 0], acc.x);    // ds_add_f32
        atomicAdd(&s_stat[base + 1], acc.y);
        atomicAdd(&s_stat[base + 2], acc.z);
        atomicAdd(&s_stat[base + 3], acc.w);
    }
    if (lane == 0) atomicAdd(&s_stat[32], ssq);
    __syncthreads();

    if (t < 33) {
        // 2048 blocks over 32 shards -> only 64 serialized adds per address.
        unsafeAtomicAdd(&ws[(blockIdx.x & (NSHARD - 1)) * 33 + t], s_stat[t]);
    }
}

// Kernel 2: combine shards; dots = means @ colsum via V_WMMA_F32_16X16X4_F32;
// per-centroid norms + argmin (first-index tie-break); write int32 index.
__global__ __launch_bounds__(256) void finish_kernel(const float* __restrict__ means,
                                                     const float* __restrict__ ws,
                                                     int* __restrict__ out) {
    __shared__ float s_stat[33];
    __shared__ float s_dots[176];
    __shared__ float s_dist[256];
    __shared__ int   s_idx[256];
    const int t = threadIdx.x;

    if (t < 33) {
        float v = 0.0f;
        #pragma unroll
        for (int s = 0; s < NSHARD; ++s) v += ws[s * 33 + t];
        s_stat[t] = v;
    }
    if (t < 176) s_dots[t] = 0.0f;
    __syncthreads();

    if (t < 32) {   // wave 0, EXEC all-ones: WMMA matvec over 11 M-tiles of 16
        const int lane = t;
        for (int mb = 0; mb < 176; mb += 16) {
            v8f c = {};
            const int r = (lane < 16) ? (mb + lane) : (mb + lane - 16);
            #pragma unroll
            for (int k = 0; k < D_DIM; k += 4) {
                // A (16x4 f32): lanes 0-15 hold K=k,k+1; lanes 16-31 hold K=k+2,k+3
                const int kk = (lane < 16) ? k : (k + 2);
                v2f a; a.x = 0.0f; a.y = 0.0f;
                if (r < L_DIM) {
                    a.x = means[r * D_DIM + kk];
                    a.y = means[r * D_DIM + kk + 1];
                }
                // B (4x16 f32): only column N=0 carries x_sum chunk
                v2f b; b.x = 0.0f; b.y = 0.0f;
                if (lane == 0)  { b.x = s_stat[k + 0]; b.y = s_stat[k + 1]; }
                if (lane == 16) { b.x = s_stat[k + 2]; b.y = s_stat[k + 3]; }
                c = __builtin_amdgcn_wmma_f32_16x16x4_f32(
                        false, a, false, b, (short)0, c, false, false);
            }
            // D column 0: lane 0 holds M=mb+v, lane 16 holds M=mb+8+v (v = VGPR 0..7)
            if (lane == 0) {
                #pragma unroll
                for (int v = 0; v < 8; ++v) s_dots[mb + v] = c[v];
            }
            if (lane == 16) {
                #pragma unroll
                for (int v = 0; v < 8; ++v) s_dots[mb + 8 + v] = c[v];
            }
        }
    }
    __syncthreads();

    const float xnorm = sqrtf(s_stat[32]);
    float dist = FLT_MAX;
    if (t < L_DIM) {
        float yn2 = 0.0f;
        #pragma unroll
        for (int d = 0; d < D_DIM; ++d) {
            const float m = means[t * D_DIM + d];
            yn2 = __fmaf_rn(m, m, yn2);
        }
        dist = 1.0f - s_dots[t] / (xnorm * sqrtf(yn2));
    }
    s_dist[t] = dist;
    s_idx[t]  = t;
    __syncthreads();

    for (int strd = 128; strd > 0; strd >>= 1) {
        if (t < strd) {
            const float d2 = s_dist[t + strd];
            const int   i2 = s_idx[t + strd];
            if (d2 < s_dist[t] || (d2 == s_dist[t] && i2 < s_idx[t])) {
                s_dist[t] = d2;
                s_idx[t]  = i2;
            }
        }
        __syncthreads();
    }
    if (t == 0) out[0] = s_idx[0];
}

extern "C" void kernel_launch(void* const* d_in, const int* in_sizes, int n_in,
                              void* d_out, int out_size, void* d_ws, size_t ws_size,
                              hipStream_t stream) {
    const float* x     = (const float*)d_in[0];   // (1<<20, 32) f32
    const float* means = (const float*)d_in[1];   // (173, 32)   f32
    float* ws = (float*)d_ws;                     // needs NSHARD*33*4 = 4224 B
    const int n4 = in_sizes[0] / 4;               // float4 count = 1<<23

    zero_ws_kernel<<<1, 256, 0, stream>>>(ws);
    stat_kernel<<<NB1, T1, 0, stream>>>(x, ws, n4);
    finish_kernel<<<1, 256, 0, stream>>>(means, ws, (int*)d_out);
}